// VQ_Quantizer_28630251995620
// MI455X (gfx1250) — compile-verified
//
#include <hip/hip_runtime.h>

// ---------------------------------------------------------------------------
// VQ quantizer for MI455X (gfx1250, wave32).
//   x:  [32, 256, 1024] f32      E: [4096, 256] f32
//   out: quant [32,256,1024] f32 followed by 1 loss float (out_size = 8388609)
//
// Compute-bound: 2*N*K*D = 68.7 GFLOP matmul vs ~68 MB total traffic
// (3 us at 23.3 TB/s). f32 precision is required for a faithful argmin,
// so the matmul runs on V_WMMA_F32_16X16X4_F32.
// ---------------------------------------------------------------------------

#define NUM_EMBED 4096
#define DIM_EMBED 256
#define BATCH     32
#define SEQ       1024
#define NPTS      (BATCH * SEQ)              // 32768 points
#define TOTAL     (BATCH * DIM_EMBED * SEQ)  // 8388608 output elements

typedef __attribute__((ext_vector_type(2))) float v2f;
typedef __attribute__((ext_vector_type(4))) float v4f;
typedef __attribute__((ext_vector_type(8))) float v8f;

// ---------------------------------------------------------------------------
// Kernel 0: enorm[k] = sum_d E[k,d]^2.  One wave per codebook row.
// ---------------------------------------------------------------------------
__global__ void __launch_bounds__(256) vq_enorm(const float* __restrict__ E,
                                                float* __restrict__ enorm) {
  const int wave = threadIdx.x >> 5;
  const int lane = threadIdx.x & 31;
  const int row  = blockIdx.x * 8 + wave;          // grid = 512 -> 4096 rows
  const float* r = E + (size_t)row * DIM_EMBED;
  float s = 0.f;
#pragma unroll
  for (int i = 0; i < 8; ++i) {                    // 256 / 32 lanes
    float v = r[lane + i * 32];
    s += v * v;
  }
#pragma unroll
  for (int m = 16; m >= 1; m >>= 1) s += __shfl_xor(s, m, 32);
  if (lane == 0) enorm[row] = s;
}

// ---------------------------------------------------------------------------
// Kernel 1: per-point argmin over the codebook via f32 WMMA.
//
// Block = 256 threads = 8 waves sharing one staged codebook tile; each wave
// owns 16 consecutive points (rows of x_flat).  A-fragments for the whole
// D=256 reduction stay resident in 64 x v2f = 128 VGPRs.  A-matrix 16x4 f32
// layout (ISA 7.12.2):
//   lanes 0-15:  M = lane,    VGPR0 = K0, VGPR1 = K1
//   lanes 16-31: M = lane-16, VGPR0 = K2, VGPR1 = K3
// x is [B, D, L] and M maps to consecutive L positions, so A loads are
// coalesced straight from global memory - no transpose needed.
//
// Codebook columns are swept in 16-wide tiles.  The 16x256 E tile is staged
// into LDS in FRAGMENT-MAJOR layout: chunk(n,hw) = n + 16*hw holds the
// contiguous stream of that lane's B K-pairs for all 64 fragments
// (128 floats), chunk stride 132 floats (528 B: 16B-aligned, and bank
// 4*chunk mod 64 distinct across each 16-lane service group -> conflict-free
// ds_load_b128).  Each b128 load fetches TWO fragments from one base
// register + immediate offset; the stream is software-pipelined one
// 2-load group ahead so LDS latency overlaps the 4 WMMAs of the current
// group instead of serializing behind s_wait_dscnt 0x0.
// ---------------------------------------------------------------------------
#define CHUNK_STRIDE 132   // floats per (n,hw) chunk: 128 payload + 4 pad

__global__ void __launch_bounds__(256)
__attribute__((amdgpu_waves_per_eu(2, 4)))
vq_argmin(const float* __restrict__ x,
          const float* __restrict__ E,
          const float* __restrict__ enorm,
          int* __restrict__ idx) {
  __shared__ float ldsB[32 * CHUNK_STRIDE];
  __shared__ float ldsEn[16];

  const int tid  = threadIdx.x;
  const int wave = tid >> 5;
  const int lane = tid & 31;
  const int m16  = lane & 15;        // column lane within fragment
  const int hw   = lane >> 4;        // half-wave: 0 or 1
  const int koff = hw * 2;           // K-pair offset within a k=4 step

  const int rowBase = blockIdx.x * 128 + wave * 16;  // 256 blocks * 128 rows
  const int b  = rowBase >> 10;                      // batch index
  const int l0 = rowBase & 1023;                     // L position base

  // ---- load all A-fragments (x stays resident in registers) ----
  v2f a[64];
  const float* xb = x + (size_t)b * (DIM_EMBED * SEQ) + (l0 + m16);
#pragma unroll
  for (int f = 0; f < 64; ++f) {
    const float* p = xb + (size_t)(4 * f + koff) * SEQ;
    a[f].x = p[0];        // K = 4f + koff
    a[f].y = p[SEQ];      // K = 4f + koff + 1
  }

  float best[8];
  int   bidx[8];
#pragma unroll
  for (int j = 0; j < 8; ++j) { best[j] = 3.4e38f; bidx[j] = 0; }

  // per-lane contiguous B stream for this lane's (column, K-pair) role
  const float* bstream = &ldsB[(m16 + 16 * hw) * CHUNK_STRIDE];

  for (int colBase = 0; colBase < NUM_EMBED; colBase += 16) {
    // ---- stage 16x256 codebook tile into fragment-major LDS layout ----
    {
      const v2f* src = (const v2f*)(E + (size_t)colBase * DIM_EMBED);
#pragma unroll
      for (int i = 0; i < 8; ++i) {
        int u  = tid + i * 256;      // 0..2047 float2 units
        int r  = u >> 7;             // tile row (column index n) 0..15
        int c2 = u & 127;            // float2 index within the row: d = 2*c2
        v2f v  = src[(size_t)r * 128 + c2];
        // d = 2*c2 -> fragment f = c2>>1, half hw = c2&1, j = 0/1
        int off = (r + 16 * (c2 & 1)) * CHUNK_STRIDE + 2 * (c2 >> 1);
        ldsB[off]     = v.x;
        ldsB[off + 1] = v.y;
      }
      if (tid < 16) ldsEn[tid] = enorm[colBase + tid];
    }
    __syncthreads();

    // ---- 16x16 dot tile: 64 chained f32 WMMAs over D=256.
    //      One ds_load_b128 = 2 fragments; pipeline 2 loads (4 frags) ahead.
    v8f acc = {};
    v4f cur0 = *(const v4f*)&bstream[0];    // fragments 0,1
    v4f cur1 = *(const v4f*)&bstream[4];    // fragments 2,3
#pragma unroll
    for (int g = 0; g < 16; ++g) {          // 16 groups x 4 fragments
      v4f nx0, nx1;
      if (g < 15) {
        nx0 = *(const v4f*)&bstream[(g + 1) * 8];
        nx1 = *(const v4f*)&bstream[(g + 1) * 8 + 4];
      }
      v2f b0 = {cur0.x, cur0.y};
      v2f b1 = {cur0.z, cur0.w};
      v2f b2 = {cur1.x, cur1.y};
      v2f b3 = {cur1.z, cur1.w};
      acc = __builtin_amdgcn_wmma_f32_16x16x4_f32(false, a[4*g+0], false, b0,
                                                  (short)0, acc, false, false);
      acc = __builtin_amdgcn_wmma_f32_16x16x4_f32(false, a[4*g+1], false, b1,
                                                  (short)0, acc, false, false);
      acc = __builtin_amdgcn_wmma_f32_16x16x4_f32(false, a[4*g+2], false, b2,
                                                  (short)0, acc, false, false);
      acc = __builtin_amdgcn_wmma_f32_16x16x4_f32(false, a[4*g+3], false, b3,
                                                  (short)0, acc, false, false);
      cur0 = nx0; cur1 = nx1;
    }

    // ---- running argmin (C layout: lane = N column, VGPR j = M row) ----
    const float en  = ldsEn[m16];
    const int   col = colBase + m16;
#pragma unroll
    for (int j = 0; j < 8; ++j) {
      float score = en - 2.0f * acc[j];
      if (score < best[j]) { best[j] = score; bidx[j] = col; }  // first-idx ties
    }
    __syncthreads();
  }

  // ---- min-reduce across the 16 column lanes of each half-wave ----
#pragma unroll
  for (int mask = 8; mask >= 1; mask >>= 1) {
#pragma unroll
    for (int j = 0; j < 8; ++j) {
      float ob = __shfl_xor(best[j], mask, 32);
      int   oi = __shfl_xor(bidx[j], mask, 32);
      if (ob < best[j] || (ob == best[j] && oi < bidx[j])) {
        best[j] = ob; bidx[j] = oi;
      }
    }
  }
  if (m16 == 0) {
#pragma unroll
    for (int j = 0; j < 8; ++j)
      idx[rowBase + j + hw * 8] = bidx[j];   // rows M=j (hw=0), M=j+8 (hw=1)
  }
}

// ---------------------------------------------------------------------------
// Kernel 2: gather quant = E[idx] into [B,D,L] layout + per-block loss partials.
// ---------------------------------------------------------------------------
__global__ void __launch_bounds__(256) vq_gather(const float* __restrict__ x,
                                                 const float* __restrict__ E,
                                                 const int* __restrict__ idx,
                                                 float* __restrict__ out,
                                                 float* __restrict__ partial) {
  __shared__ float red[256];
  float local = 0.f;
  for (int e = blockIdx.x * 256 + threadIdx.x; e < TOTAL; e += 256 * 4096) {
    int lpos = e & 1023;
    int d    = (e >> 10) & 255;
    int bb   = e >> 18;
    int n    = (bb << 10) + lpos;          // flat point index
    int k    = idx[n];
    float q  = E[(size_t)k * DIM_EMBED + d];
    out[e]   = q;                          // straight-through forward = quant
    float diff = q - x[e];
    local += diff * diff;
  }
  red[threadIdx.x] = local;
  __syncthreads();
  for (int s = 128; s > 0; s >>= 1) {
    if (threadIdx.x < s) red[threadIdx.x] += red[threadIdx.x + s];
    __syncthreads();
  }
  if (threadIdx.x == 0) partial[blockIdx.x] = red[0];
}

// ---------------------------------------------------------------------------
// Kernel 3: final loss = (1 + 0.25) * mean(diff^2).
// ---------------------------------------------------------------------------
__global__ void __launch_bounds__(256) vq_loss(const float* __restrict__ partial,
                                               float* __restrict__ lossOut) {
  __shared__ float red[256];
  float s = 0.f;
  for (int i = threadIdx.x; i < 4096; i += 256) s += partial[i];
  red[threadIdx.x] = s;
  __syncthreads();
  for (int st = 128; st > 0; st >>= 1) {
    if (threadIdx.x < st) red[threadIdx.x] += red[threadIdx.x + st];
    __syncthreads();
  }
  if (threadIdx.x == 0) lossOut[0] = red[0] * (1.25f / (float)TOTAL);
}

// ---------------------------------------------------------------------------
extern "C" void kernel_launch(void* const* d_in, const int* in_sizes, int n_in,
                              void* d_out, int out_size, void* d_ws, size_t ws_size,
                              hipStream_t stream) {
  const float* x = (const float*)d_in[0];   // [32, 256, 1024]
  const float* E = (const float*)d_in[1];   // [4096, 256]
  float* out = (float*)d_out;               // TOTAL quant floats + 1 loss float

  // workspace layout: enorm[4096] f32 | idx[32768] i32 | partial[4096] f32
  float* enorm   = (float*)d_ws;
  int*   idx     = (int*)((char*)d_ws + 16384);
  float* partial = (float*)((char*)d_ws + 16384 + 131072);

  vq_enorm <<<512, 256, 0, stream>>>(E, enorm);
  vq_argmin<<<256, 256, 0, stream>>>(x, E, enorm, idx);
  vq_gather<<<4096, 256, 0, stream>>>(x, E, idx, out, partial);
  vq_loss  <<<1, 256, 0, stream>>>(partial, out + TOTAL);
}